// ShDictRender_42726334660943
// MI455X (gfx1250) — compile-verified
//
#include <hip/hip_runtime.h>
#include <hip/hip_bf16.h>
#include <math.h>

typedef __attribute__((ext_vector_type(2))) float v2f;
typedef __attribute__((ext_vector_type(8))) float v8f;
typedef __attribute__((ext_vector_type(4))) unsigned int u32x4;
typedef __attribute__((ext_vector_type(8))) unsigned int u32x8;

// ---- problem constants (match reference) ----
#define COARSE     32
#define FINE       4
#define NUM_ATOMS  16
#define SH_DIM     9
#define DATA_DIM   28          // SH_DIM*3 + 1
#define N_INT      768         // COARSE*3*2*FINE
#define NSAMP      767         // N_INT - 1
#define NB         256         // threads per block (= 8 wave32)
#define BRAYS      256
#define GRIDRES    128         // COARSE*FINE

static constexpr float RADIUS   = 1.3f;
static constexpr float VOXEL    = RADIUS * 2.0f / (float)GRIDRES;   // 0.0203125
static constexpr float STEP     = VOXEL * 0.5f;
static constexpr float INV_VOX  = 1.0f / VOXEL;

#define CELL_FLOATS  (NUM_ATOMS * DATA_DIM)                 // 448
#define ATOM_FLOATS  (FINE*FINE*FINE*CELL_FLOATS)           // 28672 floats = 112 KB
#define SCAN_S       (ATOM_FLOATS)                          // 256 per-thread log sums
#define SCAN_P       (ATOM_FLOATS + 256)                    // 256 intra-segment inclusive prefixes
#define SCAN_SEG     (ATOM_FLOATS + 512)                    // 16 segment offsets
#define LDS_FLOATS   (ATOM_FLOATS + 528)

extern "C" __global__ void __launch_bounds__(NB)
shdict_render_kernel(const float* __restrict__ rays_o,
                     const float* __restrict__ rays_d,
                     const float* __restrict__ grid,
                     const float* __restrict__ atoms,
                     float* __restrict__ out_rgb,
                     float* __restrict__ out_alpha,
                     float* __restrict__ out_depth,
                     float* __restrict__ out_loss)
{
    extern __shared__ float lds[];
    __shared__ float red[5];   // acc, rgb0, rgb1, rgb2, depth

    const int ray = blockIdx.x;
    const int t   = threadIdx.x;

    // ---- stage atoms dictionary (112 KB) into LDS via the Tensor Data Mover ----
    // Wave 0 programs a D# (2-group form: tensors up to 2D) and issues one
    // tensor_load_to_lds: global -> LDS DMA with no VGPR traffic, tracked by
    // TENSORcnt.  Tile = 1 row of 28672 4-byte elements.
    if ((t >> 5) == 0) {
        const unsigned long long ga = (unsigned long long)atoms;
        const unsigned lds_off =
            (unsigned)(unsigned long long)(__attribute__((address_space(3))) const char*)lds;

        u32x4 g0;
        g0[0] = 1u;                                   // count=1, user mode, no gather
        g0[1] = lds_off;                              // lds_addr (bytes)
        g0[2] = (unsigned)(ga & 0xFFFFFFFFu);         // global_addr[31:0]
        g0[3] = (unsigned)((ga >> 32) & 0x01FFFFFFu)  // global_addr[56:32]
              | (2u << 30);                           // type = 2 ("image")

        u32x8 g1;
        g1[0] = (2u << 16);                           // workgroup_mask=0, data_size=2 (4B)
        g1[1] = ((unsigned)ATOM_FLOATS & 0xFFFFu) << 16;        // tensor_dim0[15:0]
        g1[2] = (((unsigned)ATOM_FLOATS >> 16) & 0xFFFFu)       // tensor_dim0[31:16]
              | (1u << 16);                                     // tensor_dim1 = 1
        g1[3] = ((unsigned)ATOM_FLOATS & 0xFFFFu) << 16;        // tile_dim0 = 28672
        g1[4] = 1u;                                             // tile_dim1 = 1, tile_dim2 = 0
        g1[5] = (unsigned)ATOM_FLOATS;                          // tensor_dim0_stride[31:0]
        g1[6] = 0u;                                             // stride0[47:32], stride1[15:0]
        g1[7] = 0u;

        asm volatile("tensor_load_to_lds %0, %1" :: "s"(g0), "s"(g1) : "memory");
        __builtin_amdgcn_s_wait_tensorcnt(0);         // TENSORcnt is per-wave
    }
    if (t < 5) red[t] = 0.0f;
    __syncthreads();                                  // publish LDS to all 8 waves

    // ---- per-ray setup ----
    const float ox = rays_o[ray*3+0], oy = rays_o[ray*3+1], oz = rays_o[ray*3+2];
    const float dx = rays_d[ray*3+0], dy = rays_d[ray*3+1], dz = rays_d[ray*3+2];
    const float nrm = sqrtf(dx*dx + dy*dy + dz*dz);
    const float start =
        fmaxf(fmaxf(fminf(( RADIUS-ox)/dx, (-RADIUS-ox)/dx),
                    fminf(( RADIUS-oy)/dy, (-RADIUS-oy)/dy)),
                    fminf(( RADIUS-oz)/dz, (-RADIUS-oz)/dz));
    const float dists = STEP * nrm;

    const float nx = dx/nrm, ny = dy/nrm, nz = dz/nrm;
    float shm[SH_DIM];
    shm[0] =  0.28209479177387814f;
    shm[1] = -0.4886025119029199f * ny;
    shm[2] =  0.4886025119029199f * nz;
    shm[3] = -0.4886025119029199f * nx;
    shm[4] =  1.0925484305920792f * nx * ny;
    shm[5] = -1.0925484305920792f * ny * nz;
    shm[6] =  0.31539156525252005f * (2.0f*nz*nz - nx*nx - ny*ny);
    shm[7] = -1.0925484305920792f * nx * nz;
    shm[8] =  0.5462742152960396f * (nx*nx - ny*ny);

    // ---- per-thread: 3 contiguous samples s = 3t..3t+2 (s=767 invalid) ----
    float alpha_j[3], q_j[3], tv_j[3], rgb_j[3][3];

    for (int j = 0; j < 3; ++j) {
        const int  s     = t*3 + j;
        const bool valid = (s < NSAMP);
        const float ts   = start + (float)s * STEP;
        tv_j[j] = ts;

        float data[DATA_DIM];
        #pragma unroll
        for (int d = 0; d < DATA_DIM; ++d) data[d] = 0.0f;

        const float px = fmaf(ts, dx, ox);
        const float py = fmaf(ts, dy, oy);
        const float pz = fmaf(ts, dz, oz);
        const bool mask = valid &&
            (px > -RADIUS) && (px < RADIUS) &&
            (py > -RADIUS) && (py < RADIUS) &&
            (pz > -RADIUS) && (pz < RADIUS);

        if (valid) {
            const float ux = (px + RADIUS) * INV_VOX;
            const float uy = (py + RADIUS) * INV_VOX;
            const float uz = (pz + RADIUS) * INV_VOX;

            for (int ci = 0; ci < 8; ++ci) {
                const float ofx = (ci & 4) ? 0.5f : -0.5f;
                const float ofy = (ci & 2) ? 0.5f : -0.5f;
                const float ofz = (ci & 1) ? 0.5f : -0.5f;
                const float fx = fminf(fmaxf(floorf(ux + ofx), 0.0f), (float)(GRIDRES-1));
                const float fy = fminf(fmaxf(floorf(uy + ofy), 0.0f), (float)(GRIDRES-1));
                const float fz = fminf(fmaxf(floorf(uz + ofz), 0.0f), (float)(GRIDRES-1));
                const float w  = (1.0f - fabsf(ux - (fx + 0.5f)))
                               * (1.0f - fabsf(uy - (fy + 0.5f)))
                               * (1.0f - fabsf(uz - (fz + 0.5f)));
                const int ix = (int)fx, iy = (int)fy, iz = (int)fz;

                const float4* cvp = (const float4*)(grid +
                    (size_t)(((ix >> 2) * COARSE + (iy >> 2)) * COARSE + (iz >> 2)) * CELL_FLOATS);
                const float4* fvp = (const float4*)(lds +
                    (((ix & 3) * FINE + (iy & 3)) * FINE + (iz & 3)) * CELL_FLOATS);

                float res[DATA_DIM];
                #pragma unroll
                for (int d = 0; d < DATA_DIM; ++d) res[d] = 0.0f;

                // 16 atoms x 28 dims: 112 global b128 + 112 ds b128 + 448 v_fma
                #pragma unroll 4
                for (int a = 0; a < NUM_ATOMS; ++a) {
                    #pragma unroll
                    for (int qq = 0; qq < 7; ++qq) {
                        const float4 cv = cvp[a*7 + qq];
                        const float4 fv = fvp[a*7 + qq];
                        res[qq*4+0] = fmaf(cv.x, fv.x, res[qq*4+0]);
                        res[qq*4+1] = fmaf(cv.y, fv.y, res[qq*4+1]);
                        res[qq*4+2] = fmaf(cv.z, fv.z, res[qq*4+2]);
                        res[qq*4+3] = fmaf(cv.w, fv.w, res[qq*4+3]);
                    }
                }
                #pragma unroll
                for (int d = 0; d < DATA_DIM; ++d) data[d] = fmaf(w, res[d], data[d]);
            }
        }

        const float sigma = mask ? fmaxf(data[DATA_DIM-1], 0.0f) : 0.0f;
        const float a     = 1.0f - expf(-sigma * dists);
        alpha_j[j] = valid ? a : 0.0f;
        q_j[j]     = valid ? (1.0f - a + 1e-10f) : 1.0f;

        #pragma unroll
        for (int c = 0; c < 3; ++c) {
            float pre = 0.0f;
            #pragma unroll
            for (int k = 0; k < SH_DIM; ++k) pre = fmaf(shm[k], data[c*SH_DIM + k], pre);
            pre = mask ? pre : 0.0f;
            rgb_j[j][c] = 1.0f / (1.0f + expf(-pre));
        }
        if (valid) out_alpha[(size_t)ray * NSAMP + s] = alpha_j[j];
    }

    // ---- block-wide exclusive scan of log-transmittance via WMMA ----
    // 256 per-thread sums S_t viewed as 16x16 matrix V[k,n] = S[n*16+k].
    // D = L * V with L lower-triangular ones (K=16 -> 4 chained f32 16x16x4 WMMAs)
    // gives the inclusive prefix within each 16-segment for all 16 segments at once.
    lds[SCAN_S + t] = logf(q_j[0] * q_j[1] * q_j[2]);
    __syncthreads();

    if (t < 32) {                      // wave 0 only; branch is wave-uniform, EXEC all-1s
        const int lane = t;
        const int n  = lane & 15;      // column (segment); also A row index m
        const int hi = lane >> 4;
        v8f acc = (v8f){0,0,0,0,0,0,0,0};
        #pragma unroll
        for (int c = 0; c < 4; ++c) {
            const int k0 = 4*c + (hi ? 2 : 0);
            const int k1 = k0 + 1;
            v2f A, Bv;
            A[0]  = (k0 <= n) ? 1.0f : 0.0f;      // lower-triangular ones
            A[1]  = (k1 <= n) ? 1.0f : 0.0f;
            Bv[0] = lds[SCAN_S + n*16 + k0];
            Bv[1] = lds[SCAN_S + n*16 + k1];
            acc = __builtin_amdgcn_wmma_f32_16x16x4_f32(
                false, A, false, Bv, (short)0, acc, false, false);
        }
        #pragma unroll
        for (int r = 0; r < 8; ++r) {             // D layout: VGPR r -> M=r (+8 for hi half)
            const int m = r + (hi ? 8 : 0);
            lds[SCAN_P + n*16 + m] = acc[r];
        }
    }
    __syncthreads();
    if (t == 0) {                                 // exclusive scan of 16 segment totals
        float run = 0.0f;
        for (int g = 0; g < 16; ++g) {
            lds[SCAN_SEG + g] = run;
            run += lds[SCAN_P + g*16 + 15];
        }
    }
    __syncthreads();

    const float E  = lds[SCAN_SEG + (t >> 4)] + lds[SCAN_P + t] - lds[SCAN_S + t];
    const float T0 = expf(E);                     // T_excl at sample 3t
    float Te[3];
    Te[0] = T0;
    Te[1] = T0 * q_j[0];
    Te[2] = Te[1] * q_j[1];

    float accA = 0.f, accR0 = 0.f, accR1 = 0.f, accR2 = 0.f, accD = 0.f;
    #pragma unroll
    for (int j = 0; j < 3; ++j) {
        const float ab = alpha_j[j] * Te[j];      // invalid sample has alpha 0
        accA  += ab;
        accR0 += ab * rgb_j[j][0];
        accR1 += ab * rgb_j[j][1];
        accR2 += ab * rgb_j[j][2];
        accD  += ab * tv_j[j];
    }
    // wave32 reduction, then one LDS atomic per wave
    #pragma unroll
    for (int off = 16; off > 0; off >>= 1) {
        accA  += __shfl_down(accA,  off, 32);
        accR0 += __shfl_down(accR0, off, 32);
        accR1 += __shfl_down(accR1, off, 32);
        accR2 += __shfl_down(accR2, off, 32);
        accD  += __shfl_down(accD,  off, 32);
    }
    if ((t & 31) == 0) {
        atomicAdd(&red[0], accA);
        atomicAdd(&red[1], accR0);
        atomicAdd(&red[2], accR1);
        atomicAdd(&red[3], accR2);
        atomicAdd(&red[4], accD);
    }
    __syncthreads();
    if (t == 0) {
        const float acc = red[0];
        out_rgb[ray*3+0] = red[1] + (1.0f - acc);
        out_rgb[ray*3+1] = red[2] + (1.0f - acc);
        out_rgb[ray*3+2] = red[3] + (1.0f - acc);
        out_depth[ray]   = red[4];
        if (ray == 0) out_loss[0] = 0.0f;
    }
}

extern "C" void kernel_launch(void* const* d_in, const int* in_sizes, int n_in,
                              void* d_out, int out_size, void* d_ws, size_t ws_size,
                              hipStream_t stream)
{
    const float* rays_o = (const float*)d_in[0];
    const float* rays_d = (const float*)d_in[1];
    const float* grid   = (const float*)d_in[2];
    const float* atoms  = (const float*)d_in[3];
    // d_in[4] = grid_id (unused)

    float* out        = (float*)d_out;
    float* out_rgb    = out;                              // 256*3
    float* out_alpha  = out + BRAYS*3;                    // 256*767
    float* out_depth  = out_alpha + (size_t)BRAYS*NSAMP;  // 256
    float* out_loss   = out_depth + BRAYS;                // 1

    const size_t lds_bytes = (size_t)LDS_FLOATS * sizeof(float);  // ~114 KB
    hipFuncSetAttribute((const void*)shdict_render_kernel,
                        hipFuncAttributeMaxDynamicSharedMemorySize, (int)lds_bytes);

    shdict_render_kernel<<<BRAYS, NB, lds_bytes, stream>>>(
        rays_o, rays_d, grid, atoms, out_rgb, out_alpha, out_depth, out_loss);
}